// SpectralBlock_76845554860774
// MI455X (gfx1250) — compile-verified
//
#include <hip/hip_runtime.h>
#include <hip/hip_bf16.h>
#include <math.h>

// ---------------------------------------------------------------------------
// SpectralBlock on MI455X (gfx1250, wave32)
//   q = eigvecs(L) sorted by |lambda| desc (== SVD U up to column signs,
//       which cancel in this op), via parallel cyclic Jacobi.
//   C = Qs^T X            -> WMMA f32 16x16x4 GEMM (async double-buffered LDS)
//   S[o,m] = sum_i W[o,i,m] * C[m,i]  -> bandwidth-bound LDS-tiled reduce
//   Out = tanh(S Qs)      -> WMMA f32 16x16x4 GEMM (+tanh epilogue)
// Workspace use: ~50 MB (A, Qt, Qs at 16 MB each + small buffers).
// ---------------------------------------------------------------------------

#define NV      2048
#define FIN     128
#define FOUT    64
#define NPAIR   (NV / 2)
#define NSWEEPS 6

typedef __attribute__((ext_vector_type(2))) float v2f;
typedef __attribute__((ext_vector_type(8))) float v8f;
typedef int v4i __attribute__((vector_size(16)));   // matches builtin param type

#if defined(__has_builtin)
#if __has_builtin(__builtin_amdgcn_wmma_f32_16x16x4_f32)
#define HAVE_WMMA_F32 1
#endif
#if __has_builtin(__builtin_amdgcn_global_load_async_to_lds_b128) && \
    __has_builtin(__builtin_amdgcn_s_wait_asynccnt)
#define HAVE_ASYNC_LDS 1
#endif
#endif

// Global (addrspace 1) / LDS (addrspace 3) typed-pointer casts for the async
// copy builtin, whose signature is (v4i as1*, v4i as3*, imm offset, imm cpol).
#define GPTR_V4I(p) ((__attribute__((address_space(1))) v4i*)(void*)(p))
#define LPTR_V4I(p) ((__attribute__((address_space(3))) v4i*)(void*)(p))

// 16-byte global -> LDS copy. Async (ASYNCcnt-tracked, no VGPR round trip)
// when the gfx1250 builtin is available; vectorized synchronous copy else.
__device__ __forceinline__ void copy16_g2l(const float* __restrict__ g,
                                           float* __restrict__ l) {
#if defined(HAVE_ASYNC_LDS)
  __builtin_amdgcn_global_load_async_to_lds_b128(
      GPTR_V4I(const_cast<float*>(g)), LPTR_V4I(l), 0, 0);
#else
  *(float4*)l = *(const float4*)g;
#endif
}

__device__ __forceinline__ void wait_g2l() {
#if defined(HAVE_ASYNC_LDS)
  __builtin_amdgcn_s_wait_asynccnt(0);
#endif
}

// ---------------------------------------------------------------------------
// Init: A = L (working copy), Qt = I (rows of Qt are eigenvectors)
// ---------------------------------------------------------------------------
__global__ void init_kernel(const float* __restrict__ L,
                            float* __restrict__ A,
                            float* __restrict__ Qt) {
  int idx = blockIdx.x * blockDim.x + threadIdx.x;       // < NV*NV
  A[idx] = L[idx];
  int r = idx / NV, c = idx % NV;
  Qt[idx] = (r == c) ? 1.0f : 0.0f;
}

// Tournament pairing for round r: pair 0 = (NV-1, r%m); pair k = ((r+k)%m,(r-k)%m)
__device__ __forceinline__ void round_pair(int round, int k, int& p, int& q) {
  const int m = NV - 1;
  int a, b;
  if (k == 0) { a = NV - 1; b = round % m; }
  else        { a = (round + k) % m; b = (round - k + m) % m; }
  p = a < b ? a : b;
  q = a < b ? b : a;
}

// ---------------------------------------------------------------------------
// Phase 1 of a Jacobi round: compute all 1024 rotation angles from old A,
// publish them indexed by column (so the update kernel can look up the
// rotation owning any column j, plus its partner column).
// ---------------------------------------------------------------------------
__global__ void jacobi_angles(const float* __restrict__ A,
                              float* __restrict__ colC,
                              float* __restrict__ colS,
                              int* __restrict__ colP,
                              int round) {
  int k = blockIdx.x * blockDim.x + threadIdx.x;
  if (k >= NPAIR) return;
  int p, q;
  round_pair(round, k, p, q);
  float app = A[(size_t)p * NV + p];
  float aqq = A[(size_t)q * NV + q];
  float apq = A[(size_t)p * NV + q];
  float c, s;
  if (fabsf(apq) < 1e-30f) {
    c = 1.0f; s = 0.0f;
  } else {
    float tau = (aqq - app) / (2.0f * apq);
    float t = copysignf(1.0f, tau) / (fabsf(tau) + sqrtf(1.0f + tau * tau));
    c = rsqrtf(1.0f + t * t);
    s = t * c;
  }
  colC[p] = c; colS[p] = s; colP[p] = q;
  colC[q] = c; colS[q] = s; colP[q] = p;
}

// ---------------------------------------------------------------------------
// Phase 2: block k owns pair (p,q). Async-stage rows p,q of A in LDS, apply
// the two-sided rotation A' = J^T A J fully for those rows (column rotation
// uses the partner column's angle), and rotate rows p,q of Qt (Qt <- J^T Qt).
// All blocks touch disjoint rows -> race-free, fully coalesced row traffic.
// ---------------------------------------------------------------------------
__global__ void __launch_bounds__(256)
jacobi_step(float* __restrict__ A, float* __restrict__ Qt,
            const float* __restrict__ colC, const float* __restrict__ colS,
            const int* __restrict__ colP, int round) {
  __shared__ __align__(16) float rowP[NV];
  __shared__ __align__(16) float rowQ[NV];
  int p, q;
  round_pair(round, blockIdx.x, p, q);
  const float c1 = colC[p];
  const float s1 = colS[p];

  const float* gp = &A[(size_t)p * NV];
  const float* gq = &A[(size_t)q * NV];
  for (int idx = threadIdx.x; idx < NV / 4; idx += blockDim.x) {
    copy16_g2l(gp + 4 * idx, &rowP[4 * idx]);
    copy16_g2l(gq + 4 * idx, &rowQ[4 * idx]);
  }
  wait_g2l();
  __syncthreads();

  for (int j = threadIdx.x; j < NV; j += blockDim.x) {
    int   jp = colP[j];
    float c2 = colC[j];
    float s2 = colS[j];
    // Row phase B = J^T A restricted to rows p,q (only our own rotation acts)
    float bpj = c1 * rowP[j]  - s1 * rowQ[j];
    float bqj = s1 * rowP[j]  + c1 * rowQ[j];
    float bpk = c1 * rowP[jp] - s1 * rowQ[jp];
    float bqk = s1 * rowP[jp] + c1 * rowQ[jp];
    // Column phase A' = B J using the rotation that owns column j
    float npj, nqj;
    if (j < jp) {            // j is the "p" column of its pair
      npj = c2 * bpj - s2 * bpk;
      nqj = c2 * bqj - s2 * bqk;
    } else {                 // j is the "q" column of its pair
      npj = s2 * bpk + c2 * bpj;
      nqj = s2 * bqk + c2 * bqj;
    }
    A[(size_t)p * NV + j] = npj;
    A[(size_t)q * NV + j] = nqj;
    // Accumulate eigenvectors: Qt <- J^T Qt (rows p,q only; same-thread RMW)
    float vp = Qt[(size_t)p * NV + j];
    float vq = Qt[(size_t)q * NV + j];
    Qt[(size_t)p * NV + j] = c1 * vp - s1 * vq;
    Qt[(size_t)q * NV + j] = s1 * vp + c1 * vq;
  }
}

// ---------------------------------------------------------------------------
// Sort modes by |eigenvalue| descending (bitonic, one workgroup, LDS).
// ---------------------------------------------------------------------------
__global__ void __launch_bounds__(1024)
sort_kernel(const float* __restrict__ A, int* __restrict__ perm) {
  __shared__ float key[NV];
  __shared__ int   sidx[NV];
  for (int i = threadIdx.x; i < NV; i += blockDim.x) {
    key[i]  = fabsf(A[(size_t)i * NV + i]);
    sidx[i] = i;
  }
  __syncthreads();
  for (int ksz = 2; ksz <= NV; ksz <<= 1) {
    for (int j = ksz >> 1; j > 0; j >>= 1) {
      for (int i = threadIdx.x; i < NV; i += blockDim.x) {
        int ixj = i ^ j;
        if (ixj > i) {
          bool desc = ((i & ksz) == 0);
          float ki = key[i], kj = key[ixj];
          bool swap = desc ? (ki < kj) : (ki > kj);
          if (swap) {
            key[i] = kj; key[ixj] = ki;
            int t = sidx[i]; sidx[i] = sidx[ixj]; sidx[ixj] = t;
          }
        }
      }
      __syncthreads();
    }
  }
  for (int i = threadIdx.x; i < NV; i += blockDim.x) perm[i] = sidx[i];
}

__global__ void gather_kernel(const float* __restrict__ Qt,
                              const int* __restrict__ perm,
                              float* __restrict__ Qs) {
  int m = blockIdx.x;
  int src = perm[m];
  for (int v = threadIdx.x; v < NV; v += blockDim.x)
    Qs[(size_t)m * NV + v] = Qt[(size_t)src * NV + v];
}

// ---------------------------------------------------------------------------
// WMMA f32 GEMM:  C[M x Nc] = A[M x K] * B[K x Nc]  (row-major)
// 8 waves/WG as WM x WN tiles of 16x16; K staged in 32-wide chunks, double
// buffered through the CDNA5 async-to-LDS path (issue next chunk, compute
// current, s_wait_asynccnt, barrier, swap).
// A-fragment layout (V_WMMA_F32_16X16X4_F32): lanes 0-15 M=0..15 K={0,1},
// lanes 16-31 M=0..15 K={2,3}. B mirrored. C/D: vgpr r, lane l ->
// (M = r + 8*(l>=16), N = l%16).
// LDS A pitch = 36 floats: 144 B (16B-aligned packets) and 36*l mod 64 banks
// are distinct across a half-wave -> conflict-free fragment reads.
// ---------------------------------------------------------------------------
template <int WM, int WN, bool TANH>
__global__ void __launch_bounds__(256)
wmma_gemm(const float* __restrict__ A, const float* __restrict__ B,
          float* __restrict__ C, int M, int Nc, int K) {
  constexpr int KC = 32;
  constexpr int AP = KC + 4;          // 36-float pitch
  constexpr int BW = WN * 16;         // B tile width (64/128 B rows)
  __shared__ __align__(16) float lA[2][WM * 16 * AP];
  __shared__ __align__(16) float lB[2][KC * BW];
  const int n0 = blockIdx.x * BW;
  const int m0 = blockIdx.y * (WM * 16);
  const int tid  = threadIdx.x;
  const int lane = tid & 31;
  const int wv   = tid >> 5;
  const int wm   = wv % WM;
  const int wn   = wv / WM;
  const int l    = lane & 15;
  const int hi   = lane >> 4;

  v8f acc = {0.f, 0.f, 0.f, 0.f, 0.f, 0.f, 0.f, 0.f};

  // Stage one K-chunk into LDS buffer `buf` (16 B packets).
  auto stage = [&](int buf, int k0) {
    for (int idx = tid; idx < WM * 16 * (KC / 4); idx += 256) {
      int r = idx >> 3, c4 = (idx & 7) * 4;              // KC/4 == 8
      copy16_g2l(&A[(size_t)(m0 + r) * K + k0 + c4], &lA[buf][r * AP + c4]);
    }
    for (int idx = tid; idx < KC * (BW / 4); idx += 256) {
      int r = idx / (BW / 4), c4 = (idx % (BW / 4)) * 4;
      copy16_g2l(&B[(size_t)(k0 + r) * Nc + n0 + c4], &lB[buf][r * BW + c4]);
    }
  };

  stage(0, 0);
  wait_g2l();
  __syncthreads();

  for (int k0 = 0, it = 0; k0 < K; k0 += KC, ++it) {
    const int cur = it & 1;
    if (k0 + KC < K) stage(cur ^ 1, k0 + KC);   // overlap next copy w/ compute

    const float* la = &lA[cur][(16 * wm + l) * AP];
    const float* lb = &lB[cur][16 * wn + l];
#pragma unroll
    for (int kk = 0; kk < KC; kk += 4) {
#if defined(HAVE_WMMA_F32)
      v2f af, bf;
      af[0] = la[kk + 2 * hi];
      af[1] = la[kk + 2 * hi + 1];
      bf[0] = lb[(kk + 2 * hi) * BW];
      bf[1] = lb[(kk + 2 * hi + 1) * BW];
      acc = __builtin_amdgcn_wmma_f32_16x16x4_f32(
          /*neg_a=*/false, af, /*neg_b=*/false, bf,
          /*c_mod=*/(short)0, acc, /*reuse_a=*/false, /*reuse_b=*/false);
#else
      // Fallback: same D-layout, computed through LDS (no lane exchange)
#pragma unroll
      for (int r = 0; r < 8; ++r) {
        float a0 = 0.f;
#pragma unroll
        for (int t = 0; t < 4; ++t)
          a0 += lA[cur][(16 * wm + r + 8 * hi) * AP + kk + t] *
                lB[cur][(kk + t) * BW + 16 * wn + l];
        acc[r] += a0;
      }
#endif
    }
    wait_g2l();          // next chunk's async copies are home
    __syncthreads();
  }

#pragma unroll
  for (int r = 0; r < 8; ++r) {
    float v = acc[r];
    if (TANH) v = tanhf(v);
    C[(size_t)(m0 + 16 * wm + r + 8 * hi) * Nc + n0 + 16 * wn + l] = v;
  }
}

// ---------------------------------------------------------------------------
// S[o,m] = sum_i W[o,i,m] * C[m,i]. 64-row tile of C in LDS (pitch 129 ->
// conflict-free per-thread row walk); weight reads are 256B coalesced bursts.
// ---------------------------------------------------------------------------
__global__ void __launch_bounds__(64)
modewise_kernel(const float* __restrict__ W, const float* __restrict__ Cm,
                float* __restrict__ S) {
  __shared__ float lc[64 * 129];
  const int o  = blockIdx.y;
  const int m0 = blockIdx.x * 64;
  for (int idx = threadIdx.x; idx < 64 * FIN; idx += 64) {
    int mm = idx >> 7, i = idx & (FIN - 1);
    lc[mm * 129 + i] = Cm[(size_t)(m0 + mm) * FIN + i];
  }
  __syncthreads();
  const int t = threadIdx.x;
  float acc = 0.f;
#pragma unroll 4
  for (int i = 0; i < FIN; ++i) {
    if ((i & 31) == 0 && i + 32 < FIN)   // hint next weight burst into cache
      __builtin_prefetch(&W[((size_t)o * FIN + i + 32) * NV + m0 + t], 0, 1);
    acc += W[((size_t)o * FIN + i) * NV + m0 + t] * lc[t * 129 + i];
  }
  S[(size_t)o * NV + m0 + t] = acc;
}

// ---------------------------------------------------------------------------
extern "C" void kernel_launch(void* const* d_in, const int* in_sizes, int n_in,
                              void* d_out, int out_size, void* d_ws, size_t ws_size,
                              hipStream_t stream) {
  (void)in_sizes; (void)n_in; (void)out_size; (void)ws_size;
  const float* L = (const float*)d_in[0];   // [2048,2048]
  const float* X = (const float*)d_in[1];   // [2048,128]
  const float* W = (const float*)d_in[2];   // [64,128,2048]
  float* out = (float*)d_out;               // [64,2048]

  float* ws   = (float*)d_ws;
  float* A    = ws;                                // 16 MB working matrix
  float* Qt   = A  + (size_t)NV * NV;              // 16 MB eigvec rows
  float* Qs   = Qt + (size_t)NV * NV;              // 16 MB sorted eigvec rows
  float* Cm   = Qs + (size_t)NV * NV;              // [2048,128] spectral coeffs
  float* S    = Cm + (size_t)NV * FIN;             // [64,2048]
  float* colC = S  + (size_t)FOUT * NV;            // per-column cos
  float* colS = colC + NV;                         // per-column sin
  int*   colP = (int*)(colS + NV);                 // per-column partner
  int*   perm = colP + NV;                         // sorted mode order

  init_kernel<<<(NV * NV) / 256, 256, 0, stream>>>(L, A, Qt);

  for (int sw = 0; sw < NSWEEPS; ++sw) {
    for (int r = 0; r < NV - 1; ++r) {
      jacobi_angles<<<NPAIR / 256, 256, 0, stream>>>(A, colC, colS, colP, r);
      jacobi_step<<<NPAIR, 256, 0, stream>>>(A, Qt, colC, colS, colP, r);
    }
  }

  sort_kernel<<<1, 1024, 0, stream>>>(A, perm);
  gather_kernel<<<NV, 256, 0, stream>>>(Qt, perm, Qs);

  // C[m,i] = sum_v Qs[m,v] * X[v,i]   (2048x128 = 2048x2048 * 2048x128)
  wmma_gemm<8, 1, false><<<dim3(FIN / 16, NV / 128), 256, 0, stream>>>(
      Qs, X, Cm, NV, FIN, NV);

  // S[o,m] = sum_i W[o,i,m] * C[m,i]
  modewise_kernel<<<dim3(NV / 64, FOUT), 64, 0, stream>>>(W, Cm, S);

  // out[o,n] = tanh( sum_m S[o,m] * Qs[m,n] )  (64x2048 = 64x2048 * 2048x2048)
  wmma_gemm<4, 2, true><<<dim3(NV / 32, FOUT / 64), 256, 0, stream>>>(
      S, Qs, out, FOUT, NV, NV);
}